// BaseLSSFPN_1889785610760
// MI455X (gfx1250) — compile-verified
//
#include <hip/hip_runtime.h>

namespace {
constexpr int kD   = 112;
constexpr int kC   = 80;
constexpr int kHW  = 16 * 44;     // 704
constexpr int kN   = 6;
constexpr int kPix = kN * kHW;    // 4224 pixels -> one block each
constexpr int kVox = 128 * 128;   // 16384 BEV cells
}

typedef __attribute__((ext_vector_type(2))) float v2f;
typedef __attribute__((ext_vector_type(8))) float v8f;

__global__ void zero_f32(float* __restrict__ p, int n) {
  int i = blockIdx.x * blockDim.x + threadIdx.x;
  if (i < n) p[i] = 0.0f;
}

// One 224-thread block (7 wave32) per pixel (n,h,w).
// Phase 1: wave0 does softmax over D=112 + stages voxel indices in LDS;
//          threads 128..207 (waves 4-6) stage the 80-channel context row.
// Phase 2: wave w owns d-tile w (7 tiles of 16 rows). Straight-line,
//          statically uniform: A fragment + 8 voxel indices loaded once,
//          then an unrolled 5-iteration c-tile loop, each iteration one
//          V_WMMA_F32_16X16X4_F32 (K=1 outer product) + 8 scatter atomics
//          per lane using 32-bit saddr-form offsets.
__launch_bounds__(224, 4)
__global__ void lss_splat_wmma(const float* __restrict__ dlog,
                               const float* __restrict__ ctx,
                               const int* __restrict__ vidx,
                               float* __restrict__ out) {
  __shared__ float s_depth[kD];
  __shared__ float s_ctx[kC];
  __shared__ int   s_idx[kD];

  const int pix  = blockIdx.x;
  const int n    = pix / kHW;
  const int hw   = pix - n * kHW;
  const int lane = threadIdx.x & 31;
  const int wave = threadIdx.x >> 5;   // 0..6

  if (wave == 0) {
    // ---- softmax over D=112 within one wave32 ----
    float v[4];
    float m = -3.0e38f;
#pragma unroll
    for (int j = 0; j < 4; ++j) {
      int d = lane + 32 * j;
      v[j] = (d < kD) ? dlog[(n * kD + d) * kHW + hw] : -3.0e38f;
      m = fmaxf(m, v[j]);
    }
#pragma unroll
    for (int off = 16; off; off >>= 1) m = fmaxf(m, __shfl_xor(m, off, 32));
    float s = 0.0f;
#pragma unroll
    for (int j = 0; j < 4; ++j) {
      int d = lane + 32 * j;
      v[j] = (d < kD) ? __expf(v[j] - m) : 0.0f;
      s += v[j];
    }
#pragma unroll
    for (int off = 16; off; off >>= 1) s += __shfl_xor(s, off, 32);
    const float inv = 1.0f / s;
#pragma unroll
    for (int j = 0; j < 4; ++j) {
      int d = lane + 32 * j;
      if (d < kD) {
        s_depth[d] = v[j] * inv;
        s_idx[d]   = vidx[(n * kD + d) * kHW + hw];
      }
    }
  } else {
    // ---- stage context row (80 channels, stride kHW) into LDS ----
    int c = (int)threadIdx.x - 128;
    if (c >= 0 && c < kC) s_ctx[c] = ctx[(n * kC + c) * kHW + hw];
  }
  __syncthreads();

  const int  l16 = lane & 15;
  const int  mhi = (lane >> 4) * 8;   // C/D layout: lanes 16-31 hold M=8..15
  const bool lo  = lane < 16;
  const int  dt  = wave;              // this wave's d-tile (0..6)

  // Per-lane voxel indices for its 8 output rows (d = dt*16 + mhi + r):
  // 8 consecutive ints -> two ds_load_b128, hoisted out of the c-tile loop.
  int vix[8];
#pragma unroll
  for (int r = 0; r < 8; ++r) vix[r] = s_idx[dt * 16 + mhi + r];

  // A: 16x4 f32 -> VGPR0 = K0 (lanes 0-15) / K2 (lanes 16-31); only K=0 live.
  const v2f A = { lo ? s_depth[dt * 16 + l16] : 0.0f, 0.0f };

#pragma unroll
  for (int ct = 0; ct < 5; ++ct) {
    // B: 4x16 f32 -> VGPR0 = K0 row in lanes 0-15; only K=0 live.
    v2f Bm = { lo ? s_ctx[ct * 16 + l16] : 0.0f, 0.0f };
    v8f acc = {};
    acc = __builtin_amdgcn_wmma_f32_16x16x4_f32(false, A, false, Bm,
                                                (short)0, acc, false, false);

    // 32-bit element offset: chan*16384 + voxel (max 80*16384 < 2^21),
    // keeps the atomics in saddr-form with a single VGPR offset.
    const unsigned base = (unsigned)(ct * 16 + l16) << 14;
#pragma unroll
    for (int r = 0; r < 8; ++r) {
      atomicAdd(out + (base + (unsigned)vix[r]), acc[r]);
    }
  }
}

extern "C" void kernel_launch(void* const* d_in, const int* in_sizes, int n_in,
                              void* d_out, int out_size, void* d_ws, size_t ws_size,
                              hipStream_t stream) {
  const float* dlog = (const float*)d_in[0];   // depth_logits [1,6,112,16,44]
  const float* ctx  = (const float*)d_in[1];   // context      [1,6,80,16,44]
  const int*   vidx = (const int*)d_in[2];     // voxel_idx    [1,6,112,16,44]
  float* out = (float*)d_out;                  // [1,80,128,128] channel-major

  zero_f32<<<(out_size + 255) / 256, 256, 0, stream>>>(out, out_size);
  lss_splat_wmma<<<kPix, 224, 0, stream>>>(dlog, ctx, vidx, out);
}